// ThreshOtsu_53266184405249
// MI455X (gfx1250) — compile-verified
//
#include <hip/hip_runtime.h>
#include <hip/hip_bf16.h>
#include <math.h>

#define NBINS 256
#define SLICE_ELEMS (1024 * 1024)
#define N_SLICES 48
#define BPS 64   // histogram blocks per slice

typedef __attribute__((ext_vector_type(2))) float v2f;
typedef __attribute__((ext_vector_type(8))) float v8f;

__device__ __forceinline__ void atomicMinF(float* a, float v) {
    if (v >= 0.f) atomicMin((int*)a, __float_as_int(v));
    else          atomicMax((unsigned int*)a, __float_as_uint(v));
}
__device__ __forceinline__ void atomicMaxF(float* a, float v) {
    if (v >= 0.f) atomicMax((int*)a, __float_as_int(v));
    else          atomicMin((unsigned int*)a, __float_as_uint(v));
}

// ---------------------------------------------------------------- init ws
__global__ void k_init(float* mnmx, unsigned* hist, float* thr) {
    int i = blockIdx.x * blockDim.x + threadIdx.x;
    if (i < N_SLICES * NBINS) hist[i] = 0u;
    if (i == 0) { mnmx[0] = INFINITY; mnmx[1] = -INFINITY; }
    if (i < N_SLICES) thr[i] = 0.f;
}

// ------------------------------------------------------- global min/max
__global__ void k_minmax(const float4* __restrict__ x, long n4, float* mnmx) {
    float mn = INFINITY, mx = -INFINITY;
    long stride = (long)gridDim.x * blockDim.x;
    for (long i = blockIdx.x * (long)blockDim.x + threadIdx.x; i < n4; i += stride) {
        float4 v = x[i];
        mn = fminf(mn, fminf(fminf(v.x, v.y), fminf(v.z, v.w)));
        mx = fmaxf(mx, fmaxf(fmaxf(v.x, v.y), fmaxf(v.z, v.w)));
    }
    __shared__ float smn[256], smx[256];
    int t = threadIdx.x;
    smn[t] = mn; smx[t] = mx;
    __syncthreads();
    for (int s = 128; s > 0; s >>= 1) {
        if (t < s) { smn[t] = fminf(smn[t], smn[t + s]); smx[t] = fmaxf(smx[t], smx[t + s]); }
        __syncthreads();
    }
    if (t == 0) { atomicMinF(&mnmx[0], smn[0]); atomicMaxF(&mnmx[1], smx[0]); }
}

// ---------------------------------------------- per-slice LDS histogram
__global__ void k_hist(const float* __restrict__ x, const float* __restrict__ mnmx,
                       unsigned* __restrict__ ghist) {
    __shared__ unsigned lh[NBINS];
    for (int i = threadIdx.x; i < NBINS; i += blockDim.x) lh[i] = 0u;
    __syncthreads();

    const float mn = mnmx[0];
    const float scale = (float)NBINS / (mnmx[1] - mn);
    const int slice = blockIdx.x / BPS;
    const int chunk = blockIdx.x % BPS;
    const int chunk4 = SLICE_ELEMS / BPS / 4;  // 4096 float4s per block
    const float4* p = (const float4*)(x + (size_t)slice * SLICE_ELEMS) + (size_t)chunk * chunk4;

    for (int i = threadIdx.x; i < chunk4; i += blockDim.x) {
        __builtin_prefetch(p + i + 2 * blockDim.x, 0, 0);  // global_prefetch_b8
        float4 v = p[i];
        int b0 = min(max((int)floorf((v.x - mn) * scale), 0), NBINS - 1);
        int b1 = min(max((int)floorf((v.y - mn) * scale), 0), NBINS - 1);
        int b2 = min(max((int)floorf((v.z - mn) * scale), 0), NBINS - 1);
        int b3 = min(max((int)floorf((v.w - mn) * scale), 0), NBINS - 1);
        atomicAdd(&lh[b0], 1u);
        atomicAdd(&lh[b1], 1u);
        atomicAdd(&lh[b2], 1u);
        atomicAdd(&lh[b3], 1u);
    }
    __syncthreads();
    for (int i = threadIdx.x; i < NBINS; i += blockDim.x)
        if (lh[i]) atomicAdd(&ghist[slice * NBINS + i], lh[i]);
}

// -------------------- per-slice Otsu: cumsums via V_WMMA_F32_16X16X4_F32
// 256-bin cumsum = (V x U) + row offsets, V[i][j]=hist[16i+j], U upper-tri ones.
// One wave32 per slice; EXEC is all-1s in the WMMA region (branch-free lanes).
__global__ __launch_bounds__(32) void k_otsu(const unsigned* __restrict__ ghist,
                                             const float* __restrict__ mnmx,
                                             float* __restrict__ thr) {
    const int slice = blockIdx.x;
    const unsigned* h = ghist + slice * NBINS;
    const int l   = threadIdx.x;
    const int row = l & 15;                 // A-matrix M index for this lane
    const int kb  = (l < 16) ? 0 : 2;       // A layout: lanes 0-15 hold K=0/1, 16-31 hold K=2/3
    const int n   = l & 15;                 // B/C/D N index for this lane

    // Preload the 8 histogram values this lane contributes to the A operands.
    float aw[8], as_[8];
#pragma unroll
    for (int k = 0; k < 4; ++k) {
        int j0 = 4 * k + kb, j1 = j0 + 1;
        float h0 = (float)h[16 * row + j0];
        float h1 = (float)h[16 * row + j1];
        aw[2 * k]      = h0;
        aw[2 * k + 1]  = h1;
        as_[2 * k]     = h0 * (float)(16 * row + j0);
        as_[2 * k + 1] = h1 * (float)(16 * row + j1);
    }

    v8f cw = {};  // cumsum(hist) partial, 16x16 f32 accumulator
    v8f cs = {};  // cumsum(hist*t) partial
#pragma unroll
    for (int k = 0; k < 4; ++k) {
        v2f a, a2, b;
        a.x  = aw[2 * k];   a.y  = aw[2 * k + 1];
        a2.x = as_[2 * k];  a2.y = as_[2 * k + 1];
        int k0 = 4 * k + kb;                    // global K row of U held by this lane
        b.x = (k0     <= n) ? 1.f : 0.f;        // U[k][n] = (k <= n)
        b.y = (k0 + 1 <= n) ? 1.f : 0.f;
        cw = __builtin_amdgcn_wmma_f32_16x16x4_f32(false, a,  false, b, (short)0, cw, false, false);
        cs = __builtin_amdgcn_wmma_f32_16x16x4_f32(false, a2, false, b, (short)0, cs, false, false);
    }

    // Spill D (16x16 intra-row prefixes) to LDS per §7.12.2 C/D layout.
    __shared__ float Pw[NBINS], Ps[NBINS];
    __shared__ float offW[17], offS[17];
#pragma unroll
    for (int v = 0; v < 8; ++v) {
        int M = v + ((l < 16) ? 0 : 8);
        Pw[M * 16 + n] = cw[v];
        Ps[M * 16 + n] = cs[v];
    }
    __syncthreads();

    if (l == 0) {  // exclusive prefix of row totals (16 rows, trivial)
        float w0 = 0.f, s0 = 0.f;
        for (int i = 0; i < 16; ++i) {
            offW[i] = w0; offS[i] = s0;
            w0 += Pw[i * 16 + 15];
            s0 += Ps[i * 16 + 15];
        }
        offW[16] = w0; offS[16] = s0;
    }
    __syncthreads();

    // Between-class variance; unnormalized counts (argmax is scale-invariant).
    const float totW = offW[16], totS = offS[16];
    float bestVar = -INFINITY;
    int   bestIdx = 0;
    for (int q = 0; q < 8; ++q) {
        int m = l * 8 + q;
        float wb = Pw[m] + offW[m >> 4];
        float sb = Ps[m] + offS[m >> 4];
        float wf = totW - wb;
        float var;
        if (wb > 0.f && wf > 0.f) {
            float d = sb / wb - (totS - sb) / wf;
            var = wb * wf * d * d;
        } else {
            var = -INFINITY;
        }
        if (var > bestVar) { bestVar = var; bestIdx = m; }  // first-max semantics
    }
    __shared__ float rv[32];
    __shared__ int   ri[32];
    rv[l] = bestVar; ri[l] = bestIdx;
    __syncthreads();
    if (l == 0) {
        float bv = rv[0]; int bi = ri[0];
        for (int i = 1; i < 32; ++i)
            if (rv[i] > bv || (rv[i] == bv && ri[i] < bi)) { bv = rv[i]; bi = ri[i]; }
        const float mn = mnmx[0];
        const float span = mnmx[1] - mn;
        thr[slice] = mn + span * (float)(bi + 1) / (float)NBINS;  // edges[t_opt+1]
    }
}

// ------------------------------------------------------ apply threshold
__global__ void k_apply(const float4* __restrict__ x, float4* __restrict__ out,
                        const float* __restrict__ thr, long n4) {
    long stride = (long)gridDim.x * blockDim.x;
    for (long i = blockIdx.x * (long)blockDim.x + threadIdx.x; i < n4; i += stride) {
        int slice = (int)((i * 4) >> 20);   // 1M elements per slice
        float t = thr[slice];
        float4 v = x[i];
        v.x = (v.x <= t) ? 0.f : v.x;
        v.y = (v.y <= t) ? 0.f : v.y;
        v.z = (v.z <= t) ? 0.f : v.z;
        v.w = (v.w <= t) ? 0.f : v.w;
        out[i] = v;
    }
}

extern "C" void kernel_launch(void* const* d_in, const int* in_sizes, int n_in,
                              void* d_out, int out_size, void* d_ws, size_t ws_size,
                              hipStream_t stream) {
    const float* x = (const float*)d_in[0];
    float* out = (float*)d_out;

    // ws layout: [0,8) min/max floats | [256, 256+48*256*4) uint hists | then 48 thresholds
    float*    mnmx = (float*)d_ws;
    unsigned* hist = (unsigned*)((char*)d_ws + 256);
    float*    thr  = (float*)((char*)d_ws + 256 + (size_t)N_SLICES * NBINS * sizeof(unsigned));

    long n  = (long)in_sizes[0];  // 48 * 1024 * 1024
    long n4 = n / 4;

    k_init<<<(N_SLICES * NBINS + 255) / 256, 256, 0, stream>>>(mnmx, hist, thr);
    k_minmax<<<8192, 256, 0, stream>>>((const float4*)x, n4, mnmx);
    k_hist<<<N_SLICES * BPS, 256, 0, stream>>>(x, mnmx, hist);
    k_otsu<<<N_SLICES, 32, 0, stream>>>(hist, mnmx, thr);
    k_apply<<<8192, 256, 0, stream>>>((const float4*)x, (float4*)out, thr, n4);
}